// PreprocessHead_7567732376310
// MI455X (gfx1250) — compile-verified
//
#include <hip/hip_runtime.h>
#include <stdint.h>

#define B_IMG   8
#define N_ANCH  250000
#define N_PRE   6000
#define N_POST  300
#define WORDS   188            // ceil(N_PRE/32)
#define N_PAD   (WORDS*32)     // 6016
#define CAND    8192
#define NBUCK   2048
#define WPB     4              // mask words (32-col tiles) per iou block; 188 = 4*47
#define CHUNK_DW (32*WORDS)    // 6016 dwords = 24 KB per NMS chunk
#define NMS_THR 0.7f
#define MIN_SZ  16.0f
#define IMG_W   1000.0f
#define IMG_H   1000.0f

typedef unsigned int uint32x4 __attribute__((ext_vector_type(4)));
typedef int          int32x8  __attribute__((ext_vector_type(8)));
typedef int          int32x4  __attribute__((ext_vector_type(4)));

__device__ __forceinline__ float4 decode_one(const float4 a, const float4 l) {
    float aw = a.z - a.x, ah = a.w - a.y;
    float ax = a.x + 0.5f * aw, ay = a.y + 0.5f * ah;
    float cx = l.x * aw + ax, cy = l.y * ah + ay;
    float w = expf(l.z) * aw, h = expf(l.w) * ah;
    float4 r;
    r.x = fminf(fmaxf(cx - 0.5f * w, 0.f), IMG_W);
    r.y = fminf(fmaxf(cy - 0.5f * h, 0.f), IMG_H);
    r.z = fminf(fmaxf(cx + 0.5f * w, 0.f), IMG_W);
    r.w = fminf(fmaxf(cy + 0.5f * h, 0.f), IMG_H);
    return r;
}

// Tensor Data Mover: 1-D tile of CHUNK_DW dwords, global -> LDS. D# per ISA §8.
__device__ __forceinline__ void tdm_load_chunk(const uint32_t* gsrc, uint32_t ldsoff) {
    unsigned long long ga = (unsigned long long)(uintptr_t)gsrc;
    uint32x4 g0;
    g0[0] = 1u;                                              // count=1, user descriptor
    g0[1] = ldsoff;                                          // lds_addr (bytes)
    g0[2] = (uint32_t)(ga & 0xFFFFFFFFu);                    // global_addr[31:0]
    g0[3] = (uint32_t)((ga >> 32) & 0x01FFFFFFu) | (2u << 30); // global_addr[56:32], type=2
    int32x8 g1;
    g1[0] = (int)(2u << 16);                                 // data_size = 4 bytes
    g1[1] = (int)((CHUNK_DW & 0xFFFFu) << 16);               // tensor_dim0[15:0]  @ bits 63:48
    g1[2] = (int)(((CHUNK_DW >> 16) & 0xFFFFu) | (1u << 16));// tensor_dim0[31:16], tensor_dim1=1
    g1[3] = (int)((CHUNK_DW & 0xFFFFu) << 16);               // tile_dim0 @ bits 127:112
    g1[4] = 1;                                               // tile_dim1=1, tile_dim2=0
    g1[5] = (int)CHUNK_DW;                                   // tensor_dim0_stride[31:0]
    g1[6] = 0; g1[7] = 0;
    int32x4 z = {0, 0, 0, 0};
#if defined(__clang_major__) && (__clang_major__ >= 23)
    int32x8 z8 = {0, 0, 0, 0, 0, 0, 0, 0};
    __builtin_amdgcn_tensor_load_to_lds(g0, g1, z, z, z8, 0);
#else
    __builtin_amdgcn_tensor_load_to_lds(g0, g1, z, z, 0);
#endif
}

// ---- Stage 1: score = sigmoid(o1-o0), decode for min-size filter, key + histogram
__global__ void score_hist_kernel(const float4* __restrict__ loc,
                                  const float2* __restrict__ obj,
                                  const float4* __restrict__ anch,
                                  uint32_t* __restrict__ keys,
                                  uint32_t* __restrict__ hist) {
    int i = blockIdx.x * 256 + threadIdx.x;
    int b = blockIdx.y;
    if (i >= N_ANCH) return;
    size_t gi = (size_t)b * N_ANCH + i;
    float2 o = obj[gi];
    float score = 1.f / (1.f + expf(o.x - o.y));       // softmax(...)[1]
    float4 bx = decode_one(anch[i], loc[gi]);
    bool valid = (bx.z - bx.x >= MIN_SZ) && (bx.w - bx.y >= MIN_SZ);
    uint32_t key = valid ? (__float_as_uint(score) | 0x80000000u) : 0u;
    keys[gi] = key;
    if (key) atomicAdd(&hist[b * NBUCK + (key >> 21)], 1u);
}

// ---- Stage 2a: per-image radix-select threshold bucket for top-N_PRE
__global__ void threshold_kernel(const uint32_t* __restrict__ hist,
                                 uint32_t* __restrict__ thr) {
    int b = threadIdx.x;
    if (b >= B_IMG) return;
    int cum = 0; uint32_t t = 0;
    for (int bk = NBUCK - 1; bk >= 0; --bk) {
        cum += hist[b * NBUCK + bk];
        if (cum >= N_PRE) { t = (uint32_t)bk; break; }
    }
    thr[b] = t;
}

// ---- Stage 2b: compact candidates (key desc, idx asc packed into u64)
__global__ void compact_kernel(const uint32_t* __restrict__ keys,
                               const uint32_t* __restrict__ thr,
                               unsigned long long* __restrict__ cand,
                               uint32_t* __restrict__ cnt) {
    int i = blockIdx.x * 256 + threadIdx.x;
    int b = blockIdx.y;
    if (i >= N_ANCH) return;
    uint32_t key = keys[(size_t)b * N_ANCH + i];
    if (key && (key >> 21) >= thr[b]) {
        uint32_t pos = atomicAdd(&cnt[b], 1u);
        if (pos < CAND)
            cand[(size_t)b * CAND + pos] =
                ((unsigned long long)key << 32) | (uint32_t)(~(uint32_t)i);
    }
}

// ---- Stage 2c: per-image bitonic sort (descending) of up to 8192 candidates in LDS
__global__ __launch_bounds__(1024)
void sort_kernel(const unsigned long long* __restrict__ cand,
                 const uint32_t* __restrict__ cnt,
                 unsigned long long* __restrict__ topPacked) {
    __shared__ unsigned long long s[CAND];   // 64 KB of the 320 KB/WGP LDS
    int b = blockIdx.x;
    uint32_t n = cnt[b]; if (n > CAND) n = CAND;
    for (int p = threadIdx.x; p < CAND; p += 1024)
        s[p] = (p < (int)n) ? cand[(size_t)b * CAND + p] : 0ull;
    for (int k = 2; k <= CAND; k <<= 1)
        for (int j = k >> 1; j > 0; j >>= 1) {
            __syncthreads();
            for (int i = threadIdx.x; i < CAND; i += 1024) {
                int l = i ^ j;
                if (l > i) {
                    unsigned long long a = s[i], c = s[l];
                    bool up = ((i & k) == 0);          // descending network
                    if (up ? (a < c) : (a > c)) { s[i] = c; s[l] = a; }
                }
            }
        }
    __syncthreads();
    for (int q = threadIdx.x; q < N_PRE; q += 1024)
        topPacked[(size_t)b * N_PRE + q] = s[q];
}

// ---- Stage 3: gather + re-decode the selected boxes (padded to N_PAD with zeros)
__global__ void gather_decode_kernel(const unsigned long long* __restrict__ topPacked,
                                     const float4* __restrict__ loc,
                                     const float4* __restrict__ anch,
                                     float4* __restrict__ selBoxes) {
    int k = blockIdx.x * 256 + threadIdx.x;
    int b = blockIdx.y;
    if (k >= N_PAD) return;
    unsigned long long p = (k < N_PRE) ? topPacked[(size_t)b * N_PRE + k] : 0ull;
    float4 bx = make_float4(0.f, 0.f, 0.f, 0.f);
    if (p) {
        uint32_t idx = ~(uint32_t)p;
        bx = decode_one(anch[idx], loc[(size_t)b * N_ANCH + idx]);
    }
    selBoxes[(size_t)b * N_PAD + k] = bx;
}

// ---- Stage 4: suppression bitmask. 128 column boxes async-staged to LDS per block;
//      each thread produces WPB mask words for its row.
__global__ void iou_mask_kernel(const float4* __restrict__ selBoxes,
                                uint32_t* __restrict__ mask) {
    int wblk = blockIdx.x;                    // 0..WORDS/WPB-1
    int i = blockIdx.y * 256 + threadIdx.x;   // row
    int b = blockIdx.z;
    int col0 = wblk * (WPB * 32);
    __shared__ float4 s_cols[WPB * 32];
    if (threadIdx.x < WPB * 32) {
        uint32_t ldsoff = (uint32_t)(uintptr_t)&s_cols[threadIdx.x];
        unsigned long long gaddr = (unsigned long long)(uintptr_t)
            (selBoxes + ((size_t)b * N_PAD + col0 + threadIdx.x));
        asm volatile("global_load_async_to_lds_b128 %0, %1, off"
                     :: "v"(ldsoff), "v"(gaddr) : "memory");
    }
    asm volatile("s_wait_asynccnt 0" ::: "memory");
    __syncthreads();
    if (i >= N_PAD) return;
    float4 r = selBoxes[(size_t)b * N_PAD + i];
    float rarea = (r.z - r.x) * (r.w - r.y);
    for (int w = 0; w < WPB; ++w) {
        uint32_t m = 0u;
#pragma unroll
        for (int jj = 0; jj < 32; ++jj) {
            int j = col0 + w * 32 + jj;
            if (j > i) {
                float4 c = s_cols[w * 32 + jj];
                float ix = fminf(r.z, c.z) - fmaxf(r.x, c.x);
                float iy = fminf(r.w, c.w) - fmaxf(r.y, c.y);
                float inter = fmaxf(ix, 0.f) * fmaxf(iy, 0.f);
                float uni = rarea + (c.z - c.x) * (c.w - c.y) - inter;
                if (inter / fmaxf(uni, 1e-9f) > NMS_THR) m |= (1u << jj);
            }
        }
        mask[((size_t)b * N_PAD + i) * WORDS + wblk * WPB + w] = m;
    }
}

// ---- Stage 5: sequential greedy reduce. 24 KB chunks double-buffered via the
//      Tensor Data Mover; intra-chunk survivors resolved with broadcast LDS reads
//      (exact: mask rows only carry j>i bits, so bit r is frozen after step r).
__global__ void nms_reduce_kernel(const uint32_t* __restrict__ mask,
                                  const unsigned long long* __restrict__ topPacked,
                                  uint32_t* __restrict__ keepOut) {
    int b = blockIdx.x;
    __shared__ uint32_t s_keep[WORDS];
    __shared__ uint4 s_buf[2][CHUNK_DW / 4];   // 2 x 24 KB, 16B aligned
    for (int w = threadIdx.x; w < WORDS; w += 256) {
        uint32_t kw = 0;
        for (int t = 0; t < 32; ++t) {
            int j = w * 32 + t;
            if (j < N_PRE && topPacked[(size_t)b * N_PRE + j] != 0ull) kw |= (1u << t);
        }
        s_keep[w] = kw;
    }
    uint32_t lds0 = (uint32_t)(uintptr_t)&s_buf[0][0];
    uint32_t lds1 = (uint32_t)(uintptr_t)&s_buf[1][0];
    const uint32_t* mbase = mask + (size_t)b * N_PAD * WORDS;
    if (threadIdx.x < 32) {                       // wave 0 drives the TDM
        tdm_load_chunk(mbase, lds0);
        __builtin_amdgcn_s_wait_tensorcnt(0);
    }
    __syncthreads();
    for (int c = 0; c < WORDS; ++c) {
        if (c + 1 < WORDS && threadIdx.x < 32)    // prefetch next chunk
            tdm_load_chunk(mbase + (size_t)(c + 1) * CHUNK_DW, (c & 1) ? lds0 : lds1);
        const uint32_t* s_rows = (const uint32_t*)s_buf[c & 1];
        // intra-chunk serial survivors for word c (every thread, broadcast reads)
        uint32_t kw = s_keep[c];
#pragma unroll
        for (int r = 0; r < 32; ++r)
            if ((kw >> r) & 1u) kw &= ~s_rows[r * WORDS + c];
        __syncthreads();                          // all reads of s_keep[c] done
        for (int w = threadIdx.x; w < WORDS; w += 256) {
            uint32_t acc = 0u, kk = kw;
            while (kk) {
                int r = __ffs(kk) - 1; kk &= kk - 1;
                acc |= s_rows[r * WORDS + w];
            }
            s_keep[w] &= ~acc;                    // word c becomes exactly kw
        }
        if (threadIdx.x < 32) __builtin_amdgcn_s_wait_tensorcnt(0);
        __syncthreads();                          // next chunk ready + keep updated
    }
    for (int w = threadIdx.x; w < WORDS; w += 256)
        keepOut[b * WORDS + w] = s_keep[w];
}

// ---- Stage 6: emit first N_POST kept boxes + batch_idx + valid mask
__global__ void output_kernel(const uint32_t* __restrict__ keepOut,
                              const float4* __restrict__ selBoxes,
                              float* __restrict__ outBoxes,
                              float* __restrict__ outBidx,
                              float* __restrict__ outValid) {
    int b = blockIdx.x;
    __shared__ uint32_t s_keep[WORDS];
    __shared__ int s_sel[N_POST];
    __shared__ int s_cnt;
    for (int w = threadIdx.x; w < WORDS; w += 256) s_keep[w] = keepOut[b * WORDS + w];
    __syncthreads();
    if (threadIdx.x == 0) {
        int cnt = 0;
        for (int w = 0; w < WORDS && cnt < N_POST; ++w) {
            uint32_t kw = s_keep[w];
            while (kw && cnt < N_POST) {
                int t = __ffs(kw) - 1;
                int i = w * 32 + t;
                if (i < N_PRE) s_sel[cnt++] = i;
                kw &= kw - 1;
            }
        }
        s_cnt = cnt;
    }
    __syncthreads();
    for (int k = threadIdx.x; k < N_POST; k += 256) {
        bool valid = k < s_cnt;
        float4 bx = make_float4(0.f, 0.f, 0.f, 0.f);
        if (valid) bx = selBoxes[(size_t)b * N_PAD + s_sel[k]];
        size_t o = (size_t)b * N_POST + k;
        outBoxes[o * 4 + 0] = bx.x; outBoxes[o * 4 + 1] = bx.y;
        outBoxes[o * 4 + 2] = bx.z; outBoxes[o * 4 + 3] = bx.w;
        outBidx[o]  = (float)b;
        outValid[o] = valid ? 1.0f : 0.0f;
    }
}

extern "C" void kernel_launch(void* const* d_in, const int* in_sizes, int n_in,
                              void* d_out, int out_size, void* d_ws, size_t ws_size,
                              hipStream_t stream) {
    (void)in_sizes; (void)n_in; (void)out_size; (void)ws_size;
    const float4* loc  = (const float4*)d_in[0];   // [B,N,4] f32
    const float2* obj  = (const float2*)d_in[1];   // [B,N,2] f32
    const float4* anch = (const float4*)d_in[2];   // [N,4]   f32

    unsigned char* ws = (unsigned char*)d_ws;
    size_t off = 0;
    auto salloc = [&](size_t bytes) -> unsigned char* {
        unsigned char* p = ws + off;
        off += (bytes + 255) & ~(size_t)255;
        return p;
    };
    uint32_t*           keys      = (uint32_t*)           salloc((size_t)B_IMG * N_ANCH * 4);
    uint32_t*           hist      = (uint32_t*)           salloc((size_t)B_IMG * NBUCK * 4);
    uint32_t*           thr       = (uint32_t*)           salloc((size_t)B_IMG * 4);
    uint32_t*           cnt       = (uint32_t*)           salloc((size_t)B_IMG * 4);
    unsigned long long* cand      = (unsigned long long*) salloc((size_t)B_IMG * CAND * 8);
    unsigned long long* topPacked = (unsigned long long*) salloc((size_t)B_IMG * N_PRE * 8);
    float4*             selBoxes  = (float4*)             salloc((size_t)B_IMG * N_PAD * 16);
    uint32_t*           keepOut   = (uint32_t*)           salloc((size_t)B_IMG * WORDS * 4);
    uint32_t*           mask      = (uint32_t*)           salloc((size_t)B_IMG * N_PAD * WORDS * 4);

    hipMemsetAsync(hist, 0, (size_t)B_IMG * NBUCK * 4, stream);
    hipMemsetAsync(cnt,  0, (size_t)B_IMG * 4,        stream);

    dim3 blk(256);
    dim3 g1((N_ANCH + 255) / 256, B_IMG);
    score_hist_kernel<<<g1, blk, 0, stream>>>(loc, obj, anch, keys, hist);
    threshold_kernel<<<1, 64, 0, stream>>>(hist, thr);
    compact_kernel<<<g1, blk, 0, stream>>>(keys, thr, cand, cnt);
    sort_kernel<<<B_IMG, 1024, 0, stream>>>(cand, cnt, topPacked);
    dim3 g6((N_PAD + 255) / 256, B_IMG);
    gather_decode_kernel<<<g6, blk, 0, stream>>>(topPacked, loc, anch, selBoxes);
    dim3 g7(WORDS / WPB, (N_PAD + 255) / 256, B_IMG);
    iou_mask_kernel<<<g7, blk, 0, stream>>>(selBoxes, mask);
    nms_reduce_kernel<<<B_IMG, blk, 0, stream>>>(mask, topPacked, keepOut);

    float* outBoxes = (float*)d_out;
    float* outBidx  = outBoxes + (size_t)B_IMG * N_POST * 4;
    float* outValid = outBidx  + (size_t)B_IMG * N_POST;
    output_kernel<<<B_IMG, blk, 0, stream>>>(keepOut, selBoxes, outBoxes, outBidx, outValid);
}